// KConditionedFNO_v1_5_60833916781229
// MI455X (gfx1250) — compile-verified
//
#include <hip/hip_runtime.h>
#include <hip/hip_bf16.h>
#include <math.h>

typedef __attribute__((ext_vector_type(2))) float v2f;
typedef __attribute__((ext_vector_type(8))) float v8f;

#define HW 65536
#define CH 64

static __device__ __forceinline__ v8f wmma_f32(v2f a, v2f b, v8f c) {
  // V_WMMA_F32_16X16X4_F32 : D = A(16x4) * B(4x16) + C(16x16)
  return __builtin_amdgcn_wmma_f32_16x16x4_f32(false, a, false, b, (short)0, c,
                                               false, false);
}

static __device__ __forceinline__ float gelu_f(float x) {
  return 0.5f * x * (1.0f + erff(x * 0.70710678118654752f));
}

static __device__ __forceinline__ v2f ld2(const float* p) {
  return *(const v2f*)p;  // 8B-aligned by construction
}

// CDNA5 async Global->LDS copy (ASYNCcnt-tracked, §15.18.3 opcode 98)
static __device__ __forceinline__ void async_ld_b128(const float* lds_dst,
                                                     const float* gsrc) {
  unsigned int lds_off = (unsigned int)(unsigned long long)(const void*)lds_dst;
  unsigned long long ga = (unsigned long long)(const void*)gsrc;
  asm volatile("global_load_async_to_lds_b128 %0, %1, off"
               :: "v"(lds_off), "v"(ga)
               : "memory");
}

static __device__ __forceinline__ void wait_async0() {
  asm volatile("s_wait_asynccnt 0x0" ::: "memory");
}

// ---------------- basis tables (DFT matrices, ortho norm split 1/16 + 1/16) --
// FWf: pair-interleaved K=w: elem(w,n) at ((w>>1)*48 + n)*2 + (w&1), n<48
// FWi: pair-interleaved K=kk: elem(kk,w) at ((kk>>1)*256 + w)*2 + (kk&1)
__global__ void k_basis(float* __restrict__ FWf, float* __restrict__ FHf,
                        float* __restrict__ FHi, float* __restrict__ FWi) {
  int t = threadIdx.x;
  const float c0 = 6.283185307179586f / 256.f;
  for (int idx = t; idx < 128 * 96; idx += 256) {  // 12288
    int pair = idx / 96, rem = idx % 96;
    int n = rem >> 1, w = pair * 2 + (rem & 1);
    float v = 0.f;
    if (n < 40) {
      int ky = n >> 1;
      float th = c0 * (float)(w * ky);
      v = ((n & 1) ? -sinf(th) : cosf(th)) * 0.0625f;
    }
    FWf[idx] = v;
  }
  for (int idx = t; idx < 256 * 20; idx += 256) {
    int h = idx / 20, kx = idx % 20;
    float th = c0 * (float)(h * kx);
    float s, c;
    __sincosf(th, &s, &c);
    FHf[idx * 2] = c * 0.0625f;  FHf[idx * 2 + 1] = -s * 0.0625f;
    FHi[idx * 2] = c * 0.0625f;  FHi[idx * 2 + 1] =  s * 0.0625f;
  }
  for (int idx = t; idx < 20 * 512; idx += 256) {  // 10240
    int pair = idx / 512, rem = idx % 512;
    int w = rem >> 1, kk = pair * 2 + (rem & 1);
    int ky = kk >> 1;
    float th = c0 * (float)(ky * w);
    float amp = (ky == 0) ? 0.0625f : 0.125f;
    float v;
    if (kk & 1) v = (ky == 0) ? 0.f : -sinf(th) * amp;
    else        v = cosf(th) * amp;
    FWi[idx] = v;
  }
}

// ---------------- k-encoder MLP + FiLM (tiny, one block) ---------------------
static __device__ __forceinline__ float block_sum256(float v, float* red) {
  int t = threadIdx.x;
  red[t] = v;
  __syncthreads();
  for (int s = 128; s > 0; s >>= 1) {
    if (t < s) red[t] += red[t + s];
    __syncthreads();
  }
  float r = red[0];
  __syncthreads();
  return r;
}

__global__ void k_encoder(const float* __restrict__ k, const float* __restrict__ w1,
                          const float* __restrict__ b1, const float* __restrict__ g1,
                          const float* __restrict__ bt1, const float* __restrict__ w2,
                          const float* __restrict__ b2, const float* __restrict__ g2,
                          const float* __restrict__ bt2, const float* __restrict__ w3,
                          const float* __restrict__ b3, const float* __restrict__ fw,
                          const float* __restrict__ fb, float* __restrict__ film) {
  __shared__ float h1[128], h2[256], kf[128], red[256];
  int t = threadIdx.x;  // 256 threads
  for (int b = 0; b < 8; ++b) {
    float kb = k[b];
    if (t < 128) h1[t] = kb * w1[t] + b1[t];
    __syncthreads();
    float s = block_sum256((t < 128) ? h1[t] : 0.f, red);
    float mean = s * (1.f / 128.f);
    float vv = (t < 128) ? (h1[t] - mean) * (h1[t] - mean) : 0.f;
    float var = block_sum256(vv, red) * (1.f / 128.f);
    float inv = rsqrtf(var + 1e-5f);
    if (t < 128) h1[t] = gelu_f((h1[t] - mean) * inv * g1[t] + bt1[t]);
    __syncthreads();
    {
      float acc = b2[t];
      for (int j = 0; j < 128; ++j) acc += h1[j] * w2[j * 256 + t];
      h2[t] = acc;
    }
    __syncthreads();
    float m2 = block_sum256(h2[t], red) * (1.f / 256.f);
    float v2 = block_sum256((h2[t] - m2) * (h2[t] - m2), red) * (1.f / 256.f);
    float inv2 = rsqrtf(v2 + 1e-5f);
    h2[t] = gelu_f((h2[t] - m2) * inv2 * g2[t] + bt2[t]);
    __syncthreads();
    if (t < 128) {
      float acc = b3[t];
      for (int j = 0; j < 256; ++j) acc += h2[j] * w3[j * 128 + t];
      kf[t] = acc;
    }
    __syncthreads();
    for (int L = 0; L < 4; ++L) {
      if (t < 128) {
        float f = fb[L * 128 + t];
        for (int j = 0; j < 128; ++j) f += kf[j] * fw[L * 16384 + j * 128 + t];
        film[(L * 8 + b) * 128 + t] = f;
      }
    }
    __syncthreads();
  }
}

// ---------------- fc0 lift: x[b,c,p] = q[b,p]*w[c] + bias[c] ----------------
__global__ void k_fc0(const float* __restrict__ q, const float* __restrict__ w,
                      const float* __restrict__ bias, float* __restrict__ x) {
  long idx = (long)blockIdx.x * 256 + threadIdx.x;
  int p = (int)(idx & (HW - 1));
  int c = (int)((idx >> 16) & 63);
  int b = (int)(idx >> 22);
  x[idx] = q[(long)b * HW + p] * w[c] + bias[c];
}

// ---------------- truncated DFT along W (WMMA GEMM, K=256, N=40->48) --------
// 4 waves / block; pair-interleaved basis staged once into LDS (async copies);
// every A/B fragment is a single aligned 64-bit load.
__global__ void __launch_bounds__(128) k_dft_w(const float* __restrict__ x,
                                               const float* __restrict__ FWf,
                                               float* __restrict__ Y) {
  __shared__ float sF[128 * 96];  // 48 KB of the 320 KB WGP LDS
  int t = threadIdx.x;            // 128
  for (int j = 0; j < 24; ++j) {  // 12288 floats = 3072 b128 chunks
    int c16 = j * 128 + t;
    async_ld_b128(&sF[c16 * 4], FWf + c16 * 4);
  }
  wait_async0();
  __syncthreads();

  int lane = t & 31;
  int wave = t >> 5;
  int mtile = blockIdx.x * 4 + wave;  // 2048 blocks * 4 waves = 8192 tiles
  int row = mtile * 16 + (lane & 15);
  int khalf = (lane >> 4) * 2;
  int n = lane & 15;
  const float* xrow = x + (long)row * 256;
  v8f acc0 = {}, acc1 = {}, acc2 = {};
  for (int k0 = 0; k0 < 256; k0 += 4) {
    v2f a = ld2(xrow + k0 + khalf);
    __builtin_prefetch(xrow + k0 + 64, 0, 1);
    const float* Fp = &sF[((k0 + khalf) >> 1) * 96 + n * 2];
    v2f b0 = ld2(Fp);
    v2f b1 = ld2(Fp + 32);
    v2f b2 = ld2(Fp + 64);
    acc0 = wmma_f32(a, b0, acc0);
    acc1 = wmma_f32(a, b1, acc1);
    acc2 = wmma_f32(a, b2, acc2);
  }
  int orow = mtile * 16 + (lane >> 4) * 8;
  for (int r = 0; r < 8; ++r) {
    float* yr = Y + (long)(orow + r) * 48;
    yr[n] = acc0[r];
    yr[16 + n] = acc1[r];
    yr[32 + n] = acc2[r];
  }
}

// ---------------- truncated DFT along H (K=256, 400 modes per (b,c)) --------
__global__ void k_dft_h(const float* __restrict__ Y, const float* __restrict__ FHf,
                        float* __restrict__ Z) {
  int bc = blockIdx.x;   // 512
  int t = threadIdx.x;   // 512 (400 active)
  if (t >= 400) return;
  int kx = t / 20, ky = t % 20;
  const float* yb = Y + (long)bc * 256 * 48;
  float zr = 0.f, zi = 0.f;
  for (int h = 0; h < 256; ++h) {
    float yr = yb[h * 48 + 2 * ky], yi = yb[h * 48 + 2 * ky + 1];
    float cr = FHf[(h * 20 + kx) * 2], ci = FHf[(h * 20 + kx) * 2 + 1];
    zr += yr * cr - yi * ci;
    zi += yr * ci + yi * cr;
  }
  Z[((long)bc * 400 + t) * 2] = zr;
  Z[((long)bc * 400 + t) * 2 + 1] = zi;
}

// ---------------- complex mode mixing: U[b,i,m] = sum_c Z[b,c,m]*W[i,c,m] ---
__global__ void k_mix(const float* __restrict__ Z, const float* __restrict__ swr,
                      const float* __restrict__ swi, float* __restrict__ U) {
  int m = blockIdx.x;   // 400 modes
  int t = threadIdx.x;  // 512
  __shared__ float zr[512], zi[512];
  {
    int c = t >> 3, b = t & 7;
    zr[t] = Z[(((long)b * 64 + c) * 400 + m) * 2];
    zi[t] = Z[(((long)b * 64 + c) * 400 + m) * 2 + 1];
  }
  __syncthreads();
  int i = t >> 3, b = t & 7;
  float ur = 0.f, ui = 0.f;
  for (int c = 0; c < 64; ++c) {
    float wr = swr[((long)(i * 64 + c)) * 400 + m];
    float wi = swi[((long)(i * 64 + c)) * 400 + m];
    float xr = zr[c * 8 + b], xi = zi[c * 8 + b];
    ur += xr * wr - xi * wi;
    ui += xr * wi + xi * wr;
  }
  U[(((long)b * 64 + i) * 400 + m) * 2] = ur;
  U[(((long)b * 64 + i) * 400 + m) * 2 + 1] = ui;
}

// ---------------- inverse DFT along H: T[b,i,h,ky] = sum_kx U * e^{+} -------
__global__ void k_idft_h(const float* __restrict__ U, const float* __restrict__ FHi,
                         float* __restrict__ T) {
  long idx = (long)blockIdx.x * 256 + threadIdx.x;  // 2,621,440
  int ky = (int)(idx % 20);
  long row = idx / 20;        // (b*64+i)*256 + h
  int h = (int)(row & 255);
  long bi = row >> 8;
  const float* ub = U + bi * 800;
  float tr = 0.f, ti = 0.f;
  for (int kx = 0; kx < 20; ++kx) {
    float ur = ub[(kx * 20 + ky) * 2], ui = ub[(kx * 20 + ky) * 2 + 1];
    float cr = FHi[(h * 20 + kx) * 2], ci = FHi[(h * 20 + kx) * 2 + 1];
    tr += ur * cr - ui * ci;
    ti += ur * ci + ui * cr;
  }
  T[row * 40 + 2 * ky] = tr;
  T[row * 40 + 2 * ky + 1] = ti;
}

// ---------------- inverse real DFT along W (WMMA GEMM, K=40, N=256) ---------
// 4 waves / block: one M-tile, each wave covers 64 of 256 output columns.
// Pair-interleaved basis (40x256) staged via async Global->LDS.
__global__ void __launch_bounds__(128) k_idft_w(const float* __restrict__ T,
                                                const float* __restrict__ FWi,
                                                float* __restrict__ X1) {
  __shared__ float sF[20 * 512];  // 40 KB
  int t = threadIdx.x;            // 128
  for (int j = 0; j < 20; ++j) {  // 10240 floats = 2560 b128 chunks
    int c16 = j * 128 + t;
    async_ld_b128(&sF[c16 * 4], FWi + c16 * 4);
  }
  wait_async0();
  __syncthreads();

  int lane = t & 31;
  int nt = t >> 5;            // wave id -> 64-column slab
  int mtile = blockIdx.x;     // 8192
  int row = mtile * 16 + (lane & 15);
  int khalf = (lane >> 4) * 2;
  int n = lane & 15;
  const float* trow = T + (long)row * 40;
  v8f acc[4] = {};
  for (int k0 = 0; k0 < 40; k0 += 4) {
    v2f a = ld2(trow + k0 + khalf);
    const float* Fp = &sF[((k0 + khalf) >> 1) * 512 + (nt * 64 + n) * 2];
    for (int j = 0; j < 4; ++j) {
      v2f bm = ld2(Fp + j * 32);
      acc[j] = wmma_f32(a, bm, acc[j]);
    }
  }
  int orow = mtile * 16 + (lane >> 4) * 8;
  for (int j = 0; j < 4; ++j)
    for (int r = 0; r < 8; ++r)
      X1[(long)(orow + r) * 256 + nt * 64 + j * 16 + n] = acc[j][r];
}

// ---------------- pointwise conv + add spectral + bias (WMMA) ---------------
// S[b,o,p] = sum_c cw[o,c]*x[b,c,p] + X1[b,o,p] + cb[o]   (in-place into X1)
// 32-pixel tile per wave: A (weights) reused across two B tiles.
__global__ void k_conv_add(const float* __restrict__ x, const float* __restrict__ cwL,
                           const float* __restrict__ cbL, float* __restrict__ S) {
  int lane = threadIdx.x;
  int blk = blockIdx.x;       // 8 b * 2048 p-tiles(32)
  int b = blk >> 11;
  int p0 = (blk & 2047) * 32;
  const float* xb = x + (long)b * CH * HW;
  float* sb = S + (long)b * CH * HW;
  int khalf = (lane >> 4) * 2;
  int n = lane & 15;
  v8f acc[4][2] = {};
  for (int k0 = 0; k0 < 64; k0 += 4) {
    const float* xk0 = xb + (long)(k0 + khalf) * HW + p0 + n;
    const float* xk1 = xb + (long)(k0 + khalf + 1) * HW + p0 + n;
    v2f bm0; bm0.x = xk0[0];  bm0.y = xk1[0];
    v2f bm1; bm1.x = xk0[16]; bm1.y = xk1[16];
    __builtin_prefetch(xb + (long)(k0 + khalf + 4) * HW + p0 + n, 0, 1);
    for (int ot = 0; ot < 4; ++ot) {
      v2f a = ld2(cwL + (ot * 16 + n) * 64 + k0 + khalf);
      acc[ot][0] = wmma_f32(a, bm0, acc[ot][0]);
      acc[ot][1] = wmma_f32(a, bm1, acc[ot][1]);
    }
  }
  for (int ot = 0; ot < 4; ++ot) {
    int ob = ot * 16 + (lane >> 4) * 8;
    for (int r = 0; r < 8; ++r) {
      int o = ob + r;
      float* sp = sb + (long)o * HW + p0 + n;
      float cbo = cbL[o];
      sp[0]  = acc[ot][0][r] + sp[0]  + cbo;
      sp[16] = acc[ot][1][r] + sp[16] + cbo;
    }
  }
}

// ---------------- groupnorm stats, two-stage deterministic ------------------
__global__ void k_stats_partial(const float* __restrict__ S, float* __restrict__ part) {
  int blk = blockIdx.x;   // 64 groups * 64 chunks
  int bg = blk >> 6;
  int chunk = blk & 63;
  int t = threadIdx.x;    // 256
  const float* base = S + (long)bg * 524288 + chunk * 8192;
  float s = 0.f, s2 = 0.f;
  for (int kk = 0; kk < 32; ++kk) {
    float v = base[kk * 256 + t];
    s += v;
    s2 += v * v;
  }
  __shared__ float r1[256], r2[256];
  r1[t] = s; r2[t] = s2;
  __syncthreads();
  for (int st = 128; st > 0; st >>= 1) {
    if (t < st) { r1[t] += r1[t + st]; r2[t] += r2[t + st]; }
    __syncthreads();
  }
  if (t == 0) {
    part[blk * 2] = r1[0];
    part[blk * 2 + 1] = r2[0];
  }
}

__global__ void k_stats_final(const float* __restrict__ part, float* __restrict__ stats) {
  int bg = threadIdx.x;  // 64
  float s = 0.f, s2 = 0.f;
  for (int c = 0; c < 64; ++c) {
    s += part[(bg * 64 + c) * 2];
    s2 += part[(bg * 64 + c) * 2 + 1];
  }
  stats[bg * 2] = s;
  stats[bg * 2 + 1] = s2;
}

// ---------------- groupnorm + gelu + FiLM -----------------------------------
__global__ void k_postnorm(const float* __restrict__ S, const float* __restrict__ stats,
                           const float* __restrict__ gn_gL, const float* __restrict__ gn_bL,
                           const float* __restrict__ filmL, float* __restrict__ X) {
  long idx = (long)blockIdx.x * 256 + threadIdx.x;
  int c = (int)((idx >> 16) & 63);
  int b = (int)(idx >> 22);
  int bg = b * 8 + (c >> 3);
  float mean = stats[bg * 2] * (1.f / 524288.f);
  float var = stats[bg * 2 + 1] * (1.f / 524288.f) - mean * mean;
  float inv = rsqrtf(var + 1e-5f);
  float v = (S[idx] - mean) * inv * gn_gL[c] + gn_bL[c];
  float xb = gelu_f(v);
  float sc = filmL[b * 128 + c];
  float sh = filmL[b * 128 + 64 + c];
  X[idx] = (1.f + 0.1f * sc) * xb + 0.1f * sh;
}

// ---------------- fused head: (64->128, gelu) then dot with out_w2 ----------
__global__ void k_head(const float* __restrict__ x, const float* __restrict__ w1,
                       const float* __restrict__ b1, const float* __restrict__ w2,
                       const float* __restrict__ b2s, const float* __restrict__ gs,
                       float* __restrict__ out) {
  int lane = threadIdx.x;
  int blk = blockIdx.x;      // 8 b * 4096 p-tiles
  int b = blk >> 12;
  int p0 = (blk & 4095) * 16;
  const float* xb = x + (long)b * CH * HW;
  int khalf = (lane >> 4) * 2;
  int n = lane & 15;
  v8f acc[8] = {};
  for (int k0 = 0; k0 < 64; k0 += 4) {
    v2f bm;
    bm.x = xb[(long)(k0 + khalf) * HW + p0 + n];
    bm.y = xb[(long)(k0 + khalf + 1) * HW + p0 + n];
    for (int ot = 0; ot < 8; ++ot) {
      v2f a = ld2(w1 + (ot * 16 + n) * 64 + k0 + khalf);
      acc[ot] = wmma_f32(a, bm, acc[ot]);
    }
  }
  float partial = 0.f;
  int ob = (lane >> 4) * 8;
  for (int ot = 0; ot < 8; ++ot)
    for (int r = 0; r < 8; ++r) {
      int o = ot * 16 + ob + r;
      float tv = gelu_f(acc[ot][r] + b1[o]);
      partial += tv * w2[o];
    }
  partial += __shfl_down(partial, 16, 32);
  if (lane < 16) out[(long)b * HW + p0 + lane] = gs[0] * (partial + b2s[0]);
}

// ---------------------------------------------------------------------------
extern "C" void kernel_launch(void* const* d_in, const int* in_sizes, int n_in,
                              void* d_out, int out_size, void* d_ws, size_t ws_size,
                              hipStream_t stream) {
  const float* k      = (const float*)d_in[0];
  const float* q      = (const float*)d_in[1];
  const float* fc0_w  = (const float*)d_in[2];
  const float* fc0_b  = (const float*)d_in[3];
  const float* ke_w1  = (const float*)d_in[4];
  const float* ke_b1  = (const float*)d_in[5];
  const float* ke_g1  = (const float*)d_in[6];
  const float* ke_bt1 = (const float*)d_in[7];
  const float* ke_w2  = (const float*)d_in[8];
  const float* ke_b2  = (const float*)d_in[9];
  const float* ke_g2  = (const float*)d_in[10];
  const float* ke_bt2 = (const float*)d_in[11];
  const float* ke_w3  = (const float*)d_in[12];
  const float* ke_b3  = (const float*)d_in[13];
  const float* sw_r   = (const float*)d_in[14];
  const float* sw_i   = (const float*)d_in[15];
  const float* cw     = (const float*)d_in[16];
  const float* cb     = (const float*)d_in[17];
  const float* gn_g   = (const float*)d_in[18];
  const float* gn_b   = (const float*)d_in[19];
  const float* fw     = (const float*)d_in[20];
  const float* fb     = (const float*)d_in[21];
  const float* out_w1 = (const float*)d_in[22];
  const float* out_b1 = (const float*)d_in[23];
  const float* out_w2 = (const float*)d_in[24];
  const float* out_b2 = (const float*)d_in[25];
  const float* gscale = (const float*)d_in[26];
  float* out = (float*)d_out;
  float* ws = (float*)d_ws;

  // workspace layout (float offsets)
  float* FILM = ws + 0;          //  4*8*128
  float* PART = ws + 4096;       //  4096*2
  float* STAT = ws + 12288;      //  64*2
  float* FWF  = ws + 16384;      //  128*96 (pair-interleaved)
  float* FHF  = ws + 28672;      //  256*20*2
  float* FHI  = ws + 38912;      //  256*20*2
  float* FWI  = ws + 49152;      //  20*512 (pair-interleaved)
  float* X    = ws + 65536;      //  8*64*65536
  float* S    = X + 33554432;    //  8*64*65536  (x1, then x1+x2)
  float* Y    = S + 33554432;    //  8*64*256*48
  float* Z    = Y + 6291456;     //  8*64*400*2
  float* U    = Z + 409600;      //  8*64*400*2
  float* T    = U + 409600;      //  8*64*256*40

  k_basis<<<1, 256, 0, stream>>>(FWF, FHF, FHI, FWI);
  k_encoder<<<1, 256, 0, stream>>>(k, ke_w1, ke_b1, ke_g1, ke_bt1, ke_w2, ke_b2,
                                   ke_g2, ke_bt2, ke_w3, ke_b3, fw, fb, FILM);
  k_fc0<<<131072, 256, 0, stream>>>(q, fc0_w, fc0_b, X);

  for (int L = 0; L < 4; ++L) {
    const float* cwL = cw + (long)L * 64 * 64;
    const float* cbL = cb + (long)L * 64;
    const float* swrL = sw_r + (long)L * 64 * 64 * 400;
    const float* swiL = sw_i + (long)L * 64 * 64 * 400;
    const float* gn_gL = gn_g + (long)L * 64;
    const float* gn_bL = gn_b + (long)L * 64;
    const float* filmL = FILM + (long)L * 8 * 128;

    k_dft_w<<<2048, 128, 0, stream>>>(X, FWF, Y);
    k_dft_h<<<512, 512, 0, stream>>>(Y, FHF, Z);
    k_mix<<<400, 512, 0, stream>>>(Z, swrL, swiL, U);
    k_idft_h<<<10240, 256, 0, stream>>>(U, FHI, T);
    k_idft_w<<<8192, 128, 0, stream>>>(T, FWI, S);
    k_conv_add<<<16384, 32, 0, stream>>>(X, cwL, cbL, S);
    k_stats_partial<<<4096, 256, 0, stream>>>(S, PART);
    k_stats_final<<<1, 64, 0, stream>>>(PART, STAT);
    k_postnorm<<<131072, 256, 0, stream>>>(S, STAT, gn_gL, gn_bL, filmL, X);
  }

  k_head<<<32768, 32, 0, stream>>>(X, out_w1, out_b1, out_w2, out_b2, gscale, out);
}